// SCTConv_45045617000964
// MI455X (gfx1250) — compile-verified
//
#include <hip/hip_runtime.h>
#include <hip/hip_bf16.h>
#include <math.h>

// Only meaningful in the device pass; host pass targeting x86 has no amdgcn builtins.
#if defined(__HIP_DEVICE_COMPILE__) && !__has_builtin(__builtin_amdgcn_wmma_f32_16x16x4_f32)
#error "missing __builtin_amdgcn_wmma_f32_16x16x4_f32 on this toolchain (device pass)"
#endif

typedef __attribute__((ext_vector_type(2))) float v2f;
typedef __attribute__((ext_vector_type(8))) float v8f;

#define FDIM 128
#define NEG_SLOPE 0.01f

__device__ __forceinline__ float leaky(float v) { return v > 0.f ? v : NEG_SLOPE * v; }

// ---------------- utility ----------------
__global__ void k_fill_zero(float* __restrict__ p, size_t n) {
  size_t i = (size_t)blockIdx.x * blockDim.x + threadIdx.x;
  if (i < n) p[i] = 0.f;
}

__global__ void k_degree(float* __restrict__ deg, const int* __restrict__ col, int E) {
  int e = blockIdx.x * blockDim.x + threadIdx.x;
  if (e < E) atomicAdd(&deg[col[e]], 1.0f);
}

__global__ void k_scales(const float* __restrict__ deg, float* __restrict__ dg,
                         float* __restrict__ ds, int N) {
  int i = blockIdx.x * blockDim.x + threadIdx.x;
  if (i < N) {
    float d = deg[i];
    dg[i] = rsqrtf(d + 1.0f);   // (deg+1)^-1/2
    ds[i] = 1.0f / d;           // deg^-1 (ring edges guarantee deg >= 2)
  }
}

// ---------------- SpMM: acc[row[e],:] += scale[col[e]] * x[col[e],:] ----------------
__global__ void k_spmm_scaled(float* __restrict__ acc, const float* __restrict__ x,
                              const float* __restrict__ scale,
                              const int* __restrict__ row, const int* __restrict__ col,
                              int E) {
  size_t idx = (size_t)blockIdx.x * blockDim.x + threadIdx.x;
  size_t total = (size_t)E * (FDIM / 4);
  if (idx >= total) return;
  int e = (int)(idx >> 5);          // FDIM/4 == 32 quads per edge
  int q = (int)(idx & 31) * 4;
  int c = col[e], r = row[e];
  float s = scale[c];
  const float4 v = *(const float4*)(x + (size_t)c * FDIM + q);
  float* out = acc + (size_t)r * FDIM + q;
  atomicAdd(out + 0, s * v.x);
  atomicAdd(out + 1, s * v.y);
  atomicAdd(out + 2, s * v.z);
  atomicAdd(out + 3, s * v.w);
}

// ---------------- GCN step finalize: f = Dg*(spmm(Dg*f) + Dg*f); h = leaky(f) ----------------
__global__ void k_gcn_fin(float* __restrict__ fout, float* __restrict__ hout,
                          const float* __restrict__ acc, const float* __restrict__ fin,
                          const float* __restrict__ dg, size_t NF) {
  size_t i = (size_t)blockIdx.x * blockDim.x + threadIdx.x;
  if (i >= NF) return;
  float d = dg[i >> 7];
  float fnew = d * (acc[i] + d * fin[i]);
  fout[i] = fnew;
  hout[i] = leaky(fnew);
}

// ---------------- scattering finalize: fp = 0.5*fp + 0.5*acc; sdiff = |fp_old - fp_new| ----------------
__global__ void k_sct_fin(float* __restrict__ fout, float* __restrict__ sdiff,
                          const float* __restrict__ acc, const float* __restrict__ fin,
                          size_t NF) {
  size_t i = (size_t)blockIdx.x * blockDim.x + threadIdx.x;
  if (i >= NF) return;
  float fold = fin[i];
  float fnew = 0.5f * (fold + acc[i]);
  fout[i] = fnew;
  if (sdiff) sdiff[i] = fabsf(fold - fnew);
}

// ---------------- attention: softmax over 6 channels, weighted mean ----------------
__global__ __launch_bounds__(128) void k_attention(float* __restrict__ hp,
                                                   const float* __restrict__ X,
                                                   const float* __restrict__ hbase, size_t NF,
                                                   const float* __restrict__ avec) {
  __shared__ float red[7][128];
  __shared__ float coef[6];
  const int n = blockIdx.x;
  const int t = threadIdx.x;
  const size_t base = (size_t)n * FDIM + t;
  float xv = X[base];
  red[0][t] = (xv > 0.f ? xv : 0.f) * avec[t];
  float a2 = avec[FDIM + t];
#pragma unroll
  for (int k = 0; k < 6; ++k) {
    float hv = hbase[(size_t)k * NF + base];
    red[1 + k][t] = (hv > 0.f ? hv : 0.f) * a2;
  }
  __syncthreads();
  for (int s = 64; s > 0; s >>= 1) {
    if (t < s) {
#pragma unroll
      for (int j = 0; j < 7; ++j) red[j][t] += red[j][t + s];
    }
    __syncthreads();
  }
  if (t == 0) {
    float xterm = red[0][0];
    float e[6], mx = -1e30f;
#pragma unroll
    for (int k = 0; k < 6; ++k) { e[k] = xterm + red[1 + k][0]; mx = fmaxf(mx, e[k]); }
    float sum = 0.f;
#pragma unroll
    for (int k = 0; k < 6; ++k) { e[k] = expf(e[k] - mx); sum += e[k]; }
    float inv = 1.0f / (6.0f * sum);   // softmax weight / 6 (mean)
#pragma unroll
    for (int k = 0; k < 6; ++k) coef[k] = e[k] * inv;
  }
  __syncthreads();
  float accv = 0.f;
#pragma unroll
  for (int k = 0; k < 6; ++k) accv += coef[k] * hbase[(size_t)k * NF + base];
  hp[base] = accv;
}

// ---------------- dense layer: out = leaky(in @ W^T + bias) via V_WMMA_F32_16X16X4_F32 ----------------
// Block = 128 threads (4 waves). Each block owns a 16-row tile of `in`; wave w
// computes output column tiles w and w+4. A-tile staged in LDS, pitch 130 to
// spread 64 LDS banks (lane m -> banks 2m,2m+1).
__global__ __launch_bounds__(128) void k_gemm_leaky(float* __restrict__ out,
                                                    const float* __restrict__ in,
                                                    const float* __restrict__ W,
                                                    const float* __restrict__ bias,
                                                    int Nrows) {
  __shared__ float Ash[16 * 130];
  const int tid  = threadIdx.x;
  const int wave = tid >> 5;
  const int lane = tid & 31;
  const int m0   = blockIdx.x * 16;

  // Stage A tile (16 x 128 f32): 512 float4, 4 per thread.
  for (int i = tid; i < 16 * 32; i += 128) {
    int r = i >> 5, c = (i & 31) * 4;
    float4 v = make_float4(0.f, 0.f, 0.f, 0.f);
    if (m0 + r < Nrows) v = *(const float4*)(in + (size_t)(m0 + r) * FDIM + c);
    float* dst = &Ash[r * 130 + c];
    dst[0] = v.x; dst[1] = v.y; dst[2] = v.z; dst[3] = v.w;
  }
  __syncthreads();

  const int mrow = lane & 15;   // A: row M   (and B/C: col N)
  const int hi   = lane >> 4;   // half-wave selector

  for (int nt = wave; nt < 8; nt += 4) {
    const int ncol = nt * 16 + (lane & 15);
    v8f c;
    const float bv = bias[ncol];
#pragma unroll
    for (int r = 0; r < 8; ++r) c[r] = bv;

#pragma unroll
    for (int k0 = 0; k0 < FDIM; k0 += 4) {
      // A frag: lane(hi,m) -> A[m][k0+2*hi], A[m][k0+2*hi+1]
      const float* ap = &Ash[mrow * 130 + k0 + 2 * hi];
      v2f a; a.x = ap[0]; a.y = ap[1];
      // B frag: B[k][n] = W^T[k][ncol] = W[ncol][k]; VGPR v holds K=k0+v (lanes 0-15), K=k0+v+2 (lanes 16-31)
      const float* bp = &W[(size_t)ncol * FDIM + k0 + 2 * hi];
      v2f b; b.x = bp[0]; b.y = bp[1];
      c = __builtin_amdgcn_wmma_f32_16x16x4_f32(false, a, false, b, (short)0, c, false, false);
    }

    // C/D layout: VGPR r <-> row (r + 8*hi), col = ncol
#pragma unroll
    for (int r = 0; r < 8; ++r) {
      const int m = m0 + r + 8 * hi;
      if (m < Nrows) out[(size_t)m * FDIM + ncol] = leaky(c[r]);
    }
  }
}

// ---------------- host launch ----------------
extern "C" void kernel_launch(void* const* d_in, const int* in_sizes, int n_in,
                              void* d_out, int out_size, void* d_ws, size_t ws_size,
                              hipStream_t stream) {
  const float* X    = (const float*)d_in[0];
  const float* W1   = (const float*)d_in[1];
  const float* b1   = (const float*)d_in[2];
  const float* W2   = (const float*)d_in[3];
  const float* b2   = (const float*)d_in[4];
  const float* avec = (const float*)d_in[5];
  const int*   edge = (const int*)d_in[6];

  const int N = in_sizes[0] / FDIM;        // 50000
  const int E = in_sizes[6] / 2;           // 700000
  const int* row = edge;
  const int* col = edge + E;
  const size_t NF = (size_t)N * FDIM;

  float* ws    = (float*)d_ws;
  float* deg   = ws;                 // [N]
  float* dg    = ws + N;             // [N]
  float* ds    = ws + 2 * (size_t)N; // [N]
  float* buf_f = ws + 3 * (size_t)N; // [N,F]  diffusion state / h_prime
  float* buf_t = buf_f + NF;         // [N,F]  spmm accumulator / layer-1 out
  float* hbase = buf_t + NF;         // [6,N,F] hA,hA2,hA3,s1,s2,s3

  const int TPB = 256;
  const int gN   = (N + TPB - 1) / TPB;
  const int gE   = (E + TPB - 1) / TPB;
  const int gNF  = (int)((NF + TPB - 1) / TPB);
  const int gSp  = (int)(((size_t)E * (FDIM / 4) + TPB - 1) / TPB);
  const int gMM  = (N + 15) / 16;

  // degrees and scales
  k_fill_zero<<<gN, TPB, 0, stream>>>(deg, (size_t)N);
  k_degree<<<gE, TPB, 0, stream>>>(deg, col, E);
  k_scales<<<gN, TPB, 0, stream>>>(deg, dg, ds, N);

  // GCN diffusion: 3 steps, collect leaky(f) into hbase[0..2]
  const float* fin = X;
  for (int it = 0; it < 3; ++it) {
    k_fill_zero<<<gNF, TPB, 0, stream>>>(buf_t, NF);
    k_spmm_scaled<<<gSp, TPB, 0, stream>>>(buf_t, fin, dg, row, col, E);
    k_gcn_fin<<<gNF, TPB, 0, stream>>>(buf_f, hbase + (size_t)it * NF, buf_t, fin, dg, NF);
    fin = buf_f;
  }

  // scattering diffusion: 4 steps, emit |sct[k-1]-sct[k]| into hbase[3..5]
  fin = X;
  for (int it = 0; it < 4; ++it) {
    k_fill_zero<<<gNF, TPB, 0, stream>>>(buf_t, NF);
    k_spmm_scaled<<<gSp, TPB, 0, stream>>>(buf_t, fin, ds, row, col, E);
    float* sd = (it >= 1) ? (hbase + (size_t)(2 + it) * NF) : nullptr;
    k_sct_fin<<<gNF, TPB, 0, stream>>>(buf_f, sd, buf_t, fin, NF);
    fin = buf_f;
  }

  // attention -> h_prime into buf_f (overwrites diffusion state; safe, read-before-write per element)
  k_attention<<<N, 128, 0, stream>>>(buf_f, X, hbase, NF, avec);

  // two dense layers with fused bias + leaky-ReLU (fp32 WMMA)
  k_gemm_leaky<<<gMM, 128, 0, stream>>>(buf_t, buf_f, W1, b1, N);
  k_gemm_leaky<<<gMM, 128, 0, stream>>>((float*)d_out, buf_t, W2, b2, N);
}